// PointTransformerLayer_5617817224085
// MI455X (gfx1250) — compile-verified
//
#include <hip/hip_runtime.h>
#include <hip/hip_bf16.h>

typedef __attribute__((ext_vector_type(2))) float v2f;
typedef __attribute__((ext_vector_type(8))) float v8f;

#define WMMA_F32X4(a, b, c) \
  __builtin_amdgcn_wmma_f32_16x16x4_f32(false, (a), false, (b), (short)0, (c), false, false)

#define NPTS 4096
#define BATCH 4
#define KNN 16

// ---------------- LDS layout for fused kernel (floats) ----------------
#define OFF_WP1 0        // w_pos1 [64][4]   (3 cols + zero pad)
#define OFF_BP1 256      // b_pos1 [64]
#define OFF_WP2 320      // w_pos2 [64][68]
#define OFF_BP2 4672     // b_pos2 [64]
#define OFF_WA1 4736     // w_att1 [256][68]
#define OFF_BA1 22144    // b_att1 [256]
#define OFF_WA2 22400    // w_att2 [64][260]
#define OFF_BA2 39040    // b_att2 [64]
#define OFF_SB  39104    // per-wave x/h1 buffer [8][16][68] (col-major: [col][row])
#define OFF_TB  47808    // per-wave h tile buffer [8][16][20]
#define LDS_FLOATS 50368 // * 4 = 201472 bytes

// ======================================================================
// Kernel 1: exact top-K=16 nearest neighbours (stable on ties, like top_k)
// ======================================================================
__global__ __launch_bounds__(256) void pt_knn(const float* __restrict__ xyz,
                                              int* __restrict__ idx) {
  __shared__ float tx[256], ty[256], tz[256];
  const int tid = threadIdx.x;
  const int gp  = blockIdx.x * 256 + tid;   // 0 .. B*N-1 (block never straddles b)
  const int b   = gp >> 12;
  const int n   = gp & (NPTS - 1);

  const float qx = xyz[(b * 3 + 0) * NPTS + n];
  const float qy = xyz[(b * 3 + 1) * NPTS + n];
  const float qz = xyz[(b * 3 + 2) * NPTS + n];

  float bd[KNN];
  int   bi[KNN];
#pragma unroll
  for (int j = 0; j < KNN; ++j) { bd[j] = 3.4e38f; bi[j] = 0; }

  for (int t0 = 0; t0 < NPTS; t0 += 256) {
    __syncthreads();
    tx[tid] = xyz[(b * 3 + 0) * NPTS + t0 + tid];
    ty[tid] = xyz[(b * 3 + 1) * NPTS + t0 + tid];
    tz[tid] = xyz[(b * 3 + 2) * NPTS + t0 + tid];
    __syncthreads();
    for (int j = 0; j < 256; ++j) {
      const float dx = qx - tx[j], dy = qy - ty[j], dz = qz - tz[j];
      float d  = dx * dx + dy * dy + dz * dz;
      int   ci = t0 + j;
      if (d < bd[KNN - 1]) {           // strict < keeps earlier index on ties
#pragma unroll
        for (int s = 0; s < KNN; ++s) {
          const bool  lt = d < bd[s];
          const float od = bd[s];
          const int   oi = bi[s];
          bd[s] = lt ? d  : od;  bi[s] = lt ? ci : oi;
          d     = lt ? od : d;   ci    = lt ? oi : ci;
        }
      }
    }
  }
#pragma unroll
  for (int j = 0; j < KNN; ++j) idx[gp * KNN + j] = bi[j];
}

// ======================================================================
// Kernel 2: qkv = w_qkv(192x64) @ points, written TRANSPOSED [b][n][64]
// so per-neighbour gathers in the fused kernel are contiguous.
// ======================================================================
__global__ __launch_bounds__(192) void pt_qkv(const float* __restrict__ points,
                                              const float* __restrict__ w_qkv,
                                              float* __restrict__ qT,
                                              float* __restrict__ kT,
                                              float* __restrict__ vT) {
  __shared__ float pcol[64];
  const int bn = blockIdx.x;           // b*N + n
  const int b  = bn >> 12;
  const int n  = bn & (NPTS - 1);
  const int o  = threadIdx.x;          // 0..191
  if (o < 64) pcol[o] = points[(b * 64 + o) * NPTS + n];
  __syncthreads();
  float acc = 0.f;
#pragma unroll 8
  for (int c = 0; c < 64; ++c) acc += w_qkv[o * 64 + c] * pcol[c];
  if (o < 64)        qT[(size_t)bn * 64 + o]        = acc;
  else if (o < 128)  kT[(size_t)bn * 64 + (o - 64)] = acc;
  else               vT[(size_t)bn * 64 + (o - 128)] = acc;
}

// ======================================================================
// Kernel 3: fused per-point transformer. One point per wave32.
// All GEMMs via V_WMMA_F32_16X16X4_F32 (N dim = 16 neighbours).
// B-operands are staged in registers after one LDS transpose round-trip.
// ======================================================================
__global__ __launch_bounds__(256) void pt_fused(
    const float* __restrict__ xyz,
    const float* __restrict__ qT, const float* __restrict__ kT,
    const float* __restrict__ vT, const int* __restrict__ idx,
    const float* __restrict__ w_pos1, const float* __restrict__ b_pos1,
    const float* __restrict__ w_pos2, const float* __restrict__ b_pos2,
    const float* __restrict__ w_att1, const float* __restrict__ b_att1,
    const float* __restrict__ w_att2, const float* __restrict__ b_att2,
    float* __restrict__ out) {
  extern __shared__ float sh[];
  const int tid = threadIdx.x;

  // ---- stage weights into LDS (row-padded for bank-conflict-free A loads)
  for (int i = tid; i < 64 * 3; i += 256) sh[OFF_WP1 + (i / 3) * 4 + (i % 3)] = w_pos1[i];
  for (int i = tid; i < 64; i += 256) {
    sh[OFF_WP1 + i * 4 + 3] = 0.0f;            // zero K-pad column
    sh[OFF_BP1 + i] = b_pos1[i];
    sh[OFF_BP2 + i] = b_pos2[i];
    sh[OFF_BA2 + i] = b_att2[i];
  }
  for (int i = tid; i < 64 * 64; i += 256)  sh[OFF_WP2 + (i >> 6) * 68  + (i & 63)]  = w_pos2[i];
  for (int i = tid; i < 256 * 64; i += 256) sh[OFF_WA1 + (i >> 6) * 68  + (i & 63)]  = w_att1[i];
  for (int i = tid; i < 256; i += 256)      sh[OFF_BA1 + i] = b_att1[i];
  for (int i = tid; i < 64 * 256; i += 256) sh[OFF_WA2 + (i >> 8) * 260 + (i & 255)] = w_att2[i];
  __syncthreads();

  const int wave = tid >> 5;
  const int lane = tid & 31;
  const int col  = lane & 15;   // WMMA N index owned by this lane
  const int hi   = lane >> 4;   // half-wave: K-pair / row+8 selector
  const float himask = hi ? 0.0f : 1.0f;   // branch-free K=3 padding
  float* sb = sh + OFF_SB + wave * (16 * 68); // [col][row], rows 0..63
  float* tb = sh + OFF_TB + wave * (16 * 20); // [col][row], rows 0..15

  const int p  = blockIdx.x * 8 + wave;       // point id, 0..16383
  const int b  = p >> 12;
  const int n  = p & (NPTS - 1);
  const int nb = idx[p * KNN + col];          // neighbour for this lane's column

  const float* qrow = qT + (size_t)(b * NPTS + n) * 64;
  const float* krow = kT + (size_t)(b * NPTS + nb) * 64;
  const float* vrow = vT + (size_t)(b * NPTS + nb) * 64;
  __builtin_prefetch(krow, 0, 0);   // warm the gathers during the pos-MLP
  __builtin_prefetch(vrow, 0, 0);

  // ---- pos MLP layer 1: h1 = relu(w_pos1 @ grouped_xyz + b1), K=3 padded to 4
  v2f gB;
  gB.x = xyz[(b * 3 + 2 * hi) * NPTS + n] - xyz[(b * 3 + 2 * hi) * NPTS + nb];
  gB.y = (xyz[(b * 3 + 1) * NPTS + n] - xyz[(b * 3 + 1) * NPTS + nb]) * himask;
#pragma unroll
  for (int mt = 0; mt < 4; ++mt) {
    // pad column of w_pos1 is zero, so the pair load is unconditional
    v2f a = *(const v2f*)&sh[OFF_WP1 + (mt * 16 + col) * 4 + 2 * hi];
    v8f c;
#pragma unroll
    for (int r = 0; r < 8; ++r) c[r] = 0.0f;
    c = WMMA_F32X4(a, gB, c);
#pragma unroll
    for (int r = 0; r < 8; ++r) {
      float hv = fmaxf(c[r] + sh[OFF_BP1 + mt * 16 + r + 8 * hi], 0.0f);
      sb[col * 68 + mt * 16 + 8 * hi + r] = hv;   // B-operand layout for layer 2
    }
  }

  // stage h1 B-operand into registers (one LDS read instead of one per M-tile)
  v2f bop[16];
#pragma unroll
  for (int kk = 0; kk < 16; ++kk)
    bop[kk] = *(const v2f*)&sb[col * 68 + kk * 4 + 2 * hi];

  // ---- pos MLP layer 2: rel = w_pos2(64x64) @ h1 + b2   (kept in registers)
  v8f rel[4];
#pragma unroll
  for (int mt = 0; mt < 4; ++mt) {
    v8f c;
#pragma unroll
    for (int r = 0; r < 8; ++r) c[r] = sh[OFF_BP2 + mt * 16 + r + 8 * hi];
#pragma unroll
    for (int kk = 0; kk < 16; ++kk) {
      v2f a = *(const v2f*)&sh[OFF_WP2 + (mt * 16 + col) * 68 + kk * 4 + 2 * hi];
      c = WMMA_F32X4(a, bop[kk], c);
    }
    rel[mt] = c;
  }

  // ---- x = (q - gather(k)) + rel  -> LDS transpose -> registers (B layout)
#pragma unroll
  for (int mt = 0; mt < 4; ++mt) {
#pragma unroll
    for (int r = 0; r < 8; ++r) {
      const int d = mt * 16 + 8 * hi + r;
      sb[col * 68 + d] = qrow[d] - krow[d] + rel[mt][r];
    }
  }
#pragma unroll
  for (int kk = 0; kk < 16; ++kk)
    bop[kk] = *(const v2f*)&sb[col * 68 + kk * 4 + 2 * hi];

  // ---- att MLP: sim = w_att2 @ relu(w_att1 @ x + b1) + b2, fused tile-by-tile
  v8f sim[4];
#pragma unroll
  for (int mt = 0; mt < 4; ++mt)
#pragma unroll
    for (int r = 0; r < 8; ++r) sim[mt][r] = sh[OFF_BA2 + mt * 16 + r + 8 * hi];

  for (int t2 = 0; t2 < 16; ++t2) {         // 16 M-tiles of att1 == K-superchunks of att2
    v8f c;
#pragma unroll
    for (int r = 0; r < 8; ++r) c[r] = sh[OFF_BA1 + t2 * 16 + r + 8 * hi];
#pragma unroll
    for (int kk = 0; kk < 16; ++kk) {
      v2f a = *(const v2f*)&sh[OFF_WA1 + (t2 * 16 + col) * 68 + kk * 4 + 2 * hi];
      c = WMMA_F32X4(a, bop[kk], c);
    }
#pragma unroll
    for (int r = 0; r < 8; ++r) tb[col * 20 + 8 * hi + r] = fmaxf(c[r], 0.0f);

    // stage this h-tile's B-operand (4 chunks), then 4x4 WMMAs of att2
    v2f tbv[4];
#pragma unroll
    for (int j = 0; j < 4; ++j)
      tbv[j] = *(const v2f*)&tb[col * 20 + j * 4 + 2 * hi];
#pragma unroll
    for (int amt = 0; amt < 4; ++amt) {
#pragma unroll
      for (int j = 0; j < 4; ++j) {
        v2f a = *(const v2f*)&sh[OFF_WA2 + (amt * 16 + col) * 260 + t2 * 16 + j * 4 + 2 * hi];
        sim[amt] = WMMA_F32X4(a, tbv[j], sim[amt]);
      }
    }
  }

  // ---- softmax over K=16 (across lanes within each 16-lane half)
#pragma unroll
  for (int amt = 0; amt < 4; ++amt) {
#pragma unroll
    for (int r = 0; r < 8; ++r) {
      float s = sim[amt][r];
      float m = s;
      m = fmaxf(m, __shfl_xor(m, 1, 32));
      m = fmaxf(m, __shfl_xor(m, 2, 32));
      m = fmaxf(m, __shfl_xor(m, 4, 32));
      m = fmaxf(m, __shfl_xor(m, 8, 32));
      float e = __expf(s - m);
      float sum = e;
      sum += __shfl_xor(sum, 1, 32);
      sum += __shfl_xor(sum, 2, 32);
      sum += __shfl_xor(sum, 4, 32);
      sum += __shfl_xor(sum, 8, 32);
      sim[amt][r] = e / sum;
    }
  }

  // ---- agg = sum_k attn * (gather(v) + rel)
#pragma unroll
  for (int amt = 0; amt < 4; ++amt) {
#pragma unroll
    for (int r = 0; r < 8; ++r) {
      const int d = amt * 16 + 8 * hi + r;
      float a = sim[amt][r] * (vrow[d] + rel[amt][r]);
      a += __shfl_xor(a, 1, 32);
      a += __shfl_xor(a, 2, 32);
      a += __shfl_xor(a, 4, 32);
      a += __shfl_xor(a, 8, 32);
      if (col == 0) out[((size_t)b * 64 + d) * NPTS + n] = a;
    }
  }
}

// ======================================================================
extern "C" void kernel_launch(void* const* d_in, const int* in_sizes, int n_in,
                              void* d_out, int out_size, void* d_ws, size_t ws_size,
                              hipStream_t stream) {
  const float* xyz    = (const float*)d_in[0];
  const float* points = (const float*)d_in[1];
  const float* w_qkv  = (const float*)d_in[2];
  const float* w_pos1 = (const float*)d_in[3];
  const float* b_pos1 = (const float*)d_in[4];
  const float* w_pos2 = (const float*)d_in[5];
  const float* b_pos2 = (const float*)d_in[6];
  const float* w_att1 = (const float*)d_in[7];
  const float* b_att1 = (const float*)d_in[8];
  const float* w_att2 = (const float*)d_in[9];
  const float* b_att2 = (const float*)d_in[10];
  float* out = (float*)d_out;

  // workspace: idx (1 MB) | qT (4 MB) | kT (4 MB) | vT (4 MB)
  char* ws = (char*)d_ws;
  int*   idx = (int*)ws;
  float* qT  = (float*)(ws + (1u << 20));
  float* kT  = (float*)(ws + (5u << 20));
  float* vT  = (float*)(ws + (9u << 20));

  const int totalPts = BATCH * NPTS; // 16384

  pt_knn<<<totalPts / 256, 256, 0, stream>>>(xyz, idx);
  pt_qkv<<<totalPts, 192, 0, stream>>>(points, w_qkv, qT, kT, vT);

  hipFuncSetAttribute(reinterpret_cast<const void*>(pt_fused),
                      hipFuncAttributeMaxDynamicSharedMemorySize,
                      LDS_FLOATS * 4);
  pt_fused<<<totalPts / 8, 256, LDS_FLOATS * 4, stream>>>(
      xyz, qT, kT, vT, idx,
      w_pos1, b_pos1, w_pos2, b_pos2,
      w_att1, b_att1, w_att2, b_att2, out);
}